// GAT_15195594293540
// MI455X (gfx1250) — compile-verified
//
#include <hip/hip_runtime.h>
#include <hip/hip_bf16.h>

// ---------------------------------------------------------------------------
// GAT (2-layer) for MI455X / gfx1250.
//  - Dense node transforms on the matrix pipe: V_WMMA_F32_16X16X4_F32 (exact f32).
//  - Edge softmax: 3 passes (atomic max via ordered-uint -> global_atomic_max_u32,
//    atomic exp-sum / weighted aggregate -> global_atomic_add_f32).
//  - edge_index (int64 [2,E]) packed once into int2 (src,dst) incl. self-loops;
//    node-sized arrays (<=13MB) are L2-resident on the 192MB L2.
//  - Layer-1 max/sum passes: one thread per edge, float4/uint4 vector gathers.
// ---------------------------------------------------------------------------

typedef __attribute__((ext_vector_type(2))) float v2f;
typedef __attribute__((ext_vector_type(8))) float v8f;

#define HEADS   4
#define HID     8
#define F1      32          // HEADS*HID
#define NEG_SLOPE 0.2f
#define ENC_NEG_INF 0x007FFFFFu   // fenc(-inf)

__device__ __forceinline__ unsigned fenc(float f) {
    unsigned u = __float_as_uint(f);
    return (u & 0x80000000u) ? ~u : (u | 0x80000000u);
}
__device__ __forceinline__ float fdec(unsigned u) {
    u = (u & 0x80000000u) ? (u & 0x7fffffffu) : ~u;
    return __uint_as_float(u);
}
__device__ __forceinline__ float lrelu(float x) {
    return x >= 0.f ? x : NEG_SLOPE * x;
}
__device__ __forceinline__ float elu1(float v) {      // jax.nn.elu, alpha=1
    return v > 0.f ? v : expm1f(v);
}

// ---- one-time: pack int64 edge_index + self-loops into int2 (src,dst) -----
__global__ void gat_edge_prep(const long long* __restrict__ ei, int E,
                              int2* __restrict__ ep, long long total) {
    long long e = (long long)blockIdx.x * blockDim.x + threadIdx.x;
    if (e >= total) return;
    int2 sd;
    if (e < E) { sd.x = (int)ei[e]; sd.y = (int)ei[(long long)E + e]; }
    else       { sd.x = sd.y = (int)(e - E); }          // self-loop
    ep[e] = sd;
}

// -------------------- Layer 1: h1[N,32] = x[N,3] @ W1[3,32] ----------------
// One wave per 16-node tile; K padded 3->4; two 16-col tiles.
__global__ void gat_l1_xform(const float* __restrict__ x,
                             const float* __restrict__ W1,
                             float* __restrict__ h1, int N, int ntiles) {
    int wave = threadIdx.x >> 5;
    int lane = threadIdx.x & 31;
    int tile = blockIdx.x * 8 + wave;
    if (tile >= ntiles) return;                 // wave-uniform exit

    int m  = lane & 15;                         // row within tile
    int kh = lane >> 4;                         // 0: K=0,1  1: K=2,3
    int node = tile * 16 + m;
    int nodeL = node < N ? node : N - 1;        // clamp (loads only)
    float kmask = kh ? 0.f : 1.f;
    bool full = (tile * 16 + 16 <= N);          // wave-uniform

    v2f a;                                      // A 16x4 layout (2 VGPRs)
    a.x = x[nodeL * 3 + kh * 2];                // K = 0 or 2
    a.y = x[nodeL * 3 + 1] * kmask;             // K = 1, or pad K=3 -> 0

    #pragma unroll
    for (int ct = 0; ct < 2; ++ct) {
        int col = ct * 16 + (lane & 15);
        v2f b;                                  // B 4x16 layout (2 VGPRs)
        b.x = W1[(kh * 2) * F1 + col];          // row 0 or 2
        b.y = W1[1 * F1 + col] * kmask;         // row 1, or pad row 3 -> 0
        v8f c = {};
        c = __builtin_amdgcn_wmma_f32_16x16x4_f32(
                false, a, false, b, (short)0, c, false, false);
        int base = tile * 16 + kh * 8;          // C/D layout rows
        if (full) {
            #pragma unroll
            for (int r = 0; r < 8; ++r) h1[(base + r) * F1 + col] = c[r];
        } else {
            #pragma unroll
            for (int r = 0; r < 8; ++r)
                if (base + r < N) h1[(base + r) * F1 + col] = c[r];
        }
    }
}

// Per-node attention logits + init of m/s/agg.
__global__ void gat_l1_al_init(const float* __restrict__ h1,
                               const float* __restrict__ a_src,
                               const float* __restrict__ a_dst,
                               float* __restrict__ als, float* __restrict__ ald,
                               unsigned* __restrict__ m1, float* __restrict__ s1,
                               float* __restrict__ agg1, int N) {
    int n = blockIdx.x * blockDim.x + threadIdx.x;
    if (n >= N) return;
    const float* hr = h1 + (long long)n * F1;
    #pragma unroll
    for (int hd = 0; hd < HEADS; ++hd) {
        float as = 0.f, ad = 0.f;
        #pragma unroll
        for (int c = 0; c < HID; ++c) {
            float hv = hr[hd * HID + c];
            as += hv * a_src[hd * HID + c];
            ad += hv * a_dst[hd * HID + c];
        }
        als[n * HEADS + hd] = as;
        ald[n * HEADS + hd] = ad;
        m1[n * HEADS + hd]  = ENC_NEG_INF;
        s1[n * HEADS + hd]  = 0.f;
    }
    #pragma unroll
    for (int i = 0; i < F1; ++i) agg1[(long long)n * F1 + i] = 0.f;
}

// Pass 1: segment max. One thread per edge; float4 gathers over 4 heads.
__global__ void gat_l1_edge_max(const int2* __restrict__ ep,
                                const float* __restrict__ als,
                                const float* __restrict__ ald,
                                unsigned* __restrict__ m1, long long total) {
    long long e = (long long)blockIdx.x * blockDim.x + threadIdx.x;
    if (e >= total) return;
    int2 sd = ep[e];
    float4 as = *(const float4*)(als + (long long)sd.x * HEADS);
    float4 ad = *(const float4*)(ald + (long long)sd.y * HEADS);
    unsigned* mp = m1 + (long long)sd.y * HEADS;
    atomicMax(mp + 0, fenc(lrelu(as.x + ad.x)));
    atomicMax(mp + 1, fenc(lrelu(as.y + ad.y)));
    atomicMax(mp + 2, fenc(lrelu(as.z + ad.z)));
    atomicMax(mp + 3, fenc(lrelu(as.w + ad.w)));
}

// Pass 2: exp-sum. One thread per edge; vector gathers.
__global__ void gat_l1_edge_sum(const int2* __restrict__ ep,
                                const float* __restrict__ als,
                                const float* __restrict__ ald,
                                const unsigned* __restrict__ m1,
                                float* __restrict__ s1, long long total) {
    long long e = (long long)blockIdx.x * blockDim.x + threadIdx.x;
    if (e >= total) return;
    int2 sd = ep[e];
    float4 as = *(const float4*)(als + (long long)sd.x * HEADS);
    float4 ad = *(const float4*)(ald + (long long)sd.y * HEADS);
    uint4  mm = *(const uint4*)(m1 + (long long)sd.y * HEADS);
    float* sp = s1 + (long long)sd.y * HEADS;
    atomicAdd(sp + 0, expf(lrelu(as.x + ad.x) - fdec(mm.x)));
    atomicAdd(sp + 1, expf(lrelu(as.y + ad.y) - fdec(mm.y)));
    atomicAdd(sp + 2, expf(lrelu(as.z + ad.z) - fdec(mm.z)));
    atomicAdd(sp + 3, expf(lrelu(as.w + ad.w) - fdec(mm.w)));
}

// Pass 3: weighted aggregate. One thread per (edge, head).
__global__ void gat_l1_edge_agg(const int2* __restrict__ ep,
                                const float* __restrict__ als,
                                const float* __restrict__ ald,
                                const unsigned* __restrict__ m1,
                                const float* __restrict__ s1,
                                const float* __restrict__ h1,
                                float* __restrict__ agg1, long long total) {
    long long t = (long long)blockIdx.x * blockDim.x + threadIdx.x;
    if (t >= total) return;
    long long e = t >> 2; int hd = (int)(t & 3);
    int2 sd = ep[e];
    float ec = lrelu(als[sd.x * HEADS + hd] + ald[sd.y * HEADS + hd]);
    float alpha = expf(ec - fdec(m1[sd.y * HEADS + hd]))
                / (s1[sd.y * HEADS + hd] + 1e-16f);
    const float4* hs = (const float4*)(h1 + (long long)sd.x * F1 + hd * HID);
    float4 h0 = hs[0], h4 = hs[1];              // 8 floats, 32B aligned
    float* od = agg1 + (long long)sd.y * F1 + hd * HID;
    atomicAdd(od + 0, alpha * h0.x);
    atomicAdd(od + 1, alpha * h0.y);
    atomicAdd(od + 2, alpha * h0.z);
    atomicAdd(od + 3, alpha * h0.w);
    atomicAdd(od + 4, alpha * h4.x);
    atomicAdd(od + 5, alpha * h4.y);
    atomicAdd(od + 6, alpha * h4.z);
    atomicAdd(od + 7, alpha * h4.w);
}

// -------------------- Layer 2: t2[N,2] = elu(agg1+b1)[N,32] @ W2[32,2] -----
// bias+ELU fused into A-operand loads (each agg1 element consumed once here).
// One wave per 16-node tile; 8 chained K=4 WMMAs; only cols 0,1 are real.
__global__ void gat_l2_xform(const float* __restrict__ agg1,
                             const float* __restrict__ b1,
                             const float* __restrict__ W2,
                             float* __restrict__ t2, int N, int ntiles) {
    int wave = threadIdx.x >> 5;
    int lane = threadIdx.x & 31;
    int tile = blockIdx.x * 8 + wave;
    if (tile >= ntiles) return;

    int m  = lane & 15;
    int kh = lane >> 4;
    int node = tile * 16 + m;
    int nodeL = node < N ? node : N - 1;
    int col = lane & 15;
    float cmask = (col < 2) ? 1.f : 0.f;
    int ccol = (col < 2) ? col : 0;             // keep indices in-bounds
    bool full = (tile * 16 + 16 <= N);

    v8f c = {};
    #pragma unroll
    for (int kk = 0; kk < 8; ++kk) {
        int k0 = kk * 4 + kh * 2;
        v2f a;
        a.x = elu1(agg1[nodeL * F1 + k0]     + b1[k0]);
        a.y = elu1(agg1[nodeL * F1 + k0 + 1] + b1[k0 + 1]);
        v2f b;
        b.x = W2[k0 * 2 + ccol] * cmask;
        b.y = W2[(k0 + 1) * 2 + ccol] * cmask;
        c = __builtin_amdgcn_wmma_f32_16x16x4_f32(
                false, a, false, b, (short)0, c, false, false);
    }
    if (col < 2) {
        int base = tile * 16 + kh * 8;
        if (full) {
            #pragma unroll
            for (int r = 0; r < 8; ++r) t2[(base + r) * 2 + col] = c[r];
        } else {
            #pragma unroll
            for (int r = 0; r < 8; ++r)
                if (base + r < N) t2[(base + r) * 2 + col] = c[r];
        }
    }
}

__global__ void gat_l2_al_init(const float* __restrict__ t2,
                               const float* __restrict__ a_src,
                               const float* __restrict__ a_dst,
                               float* __restrict__ als, float* __restrict__ ald,
                               unsigned* __restrict__ m2, float* __restrict__ s2,
                               float* __restrict__ agg2, int N) {
    int n = blockIdx.x * blockDim.x + threadIdx.x;
    if (n >= N) return;
    float t0 = t2[n * 2], t1 = t2[n * 2 + 1];
    als[n] = t0 * a_src[0] + t1 * a_src[1];
    ald[n] = t0 * a_dst[0] + t1 * a_dst[1];
    m2[n] = ENC_NEG_INF;
    s2[n] = 0.f;
    agg2[n * 2] = 0.f; agg2[n * 2 + 1] = 0.f;
}

__global__ void gat_l2_edge_max(const int2* __restrict__ ep,
                                const float* __restrict__ als,
                                const float* __restrict__ ald,
                                unsigned* __restrict__ m2, long long total) {
    long long e = (long long)blockIdx.x * blockDim.x + threadIdx.x;
    if (e >= total) return;
    int2 sd = ep[e];
    float ec = lrelu(als[sd.x] + ald[sd.y]);
    atomicMax(&m2[sd.y], fenc(ec));
}

__global__ void gat_l2_edge_sum(const int2* __restrict__ ep,
                                const float* __restrict__ als,
                                const float* __restrict__ ald,
                                const unsigned* __restrict__ m2,
                                float* __restrict__ s2, long long total) {
    long long e = (long long)blockIdx.x * blockDim.x + threadIdx.x;
    if (e >= total) return;
    int2 sd = ep[e];
    float ec = lrelu(als[sd.x] + ald[sd.y]);
    atomicAdd(&s2[sd.y], expf(ec - fdec(m2[sd.y])));
}

__global__ void gat_l2_edge_agg(const int2* __restrict__ ep,
                                const float* __restrict__ als,
                                const float* __restrict__ ald,
                                const unsigned* __restrict__ m2,
                                const float* __restrict__ s2,
                                const float* __restrict__ t2,
                                float* __restrict__ agg2, long long total) {
    long long e = (long long)blockIdx.x * blockDim.x + threadIdx.x;
    if (e >= total) return;
    int2 sd = ep[e];
    float ec = lrelu(als[sd.x] + ald[sd.y]);
    float alpha = expf(ec - fdec(m2[sd.y])) / (s2[sd.y] + 1e-16f);
    atomicAdd(&agg2[sd.y * 2],     alpha * t2[sd.x * 2]);
    atomicAdd(&agg2[sd.y * 2 + 1], alpha * t2[sd.x * 2 + 1]);
}

__global__ void gat_l2_fin(const float* __restrict__ agg2,
                           const float* __restrict__ b2,
                           float* __restrict__ out, long long total) {
    long long i = (long long)blockIdx.x * blockDim.x + threadIdx.x;
    if (i >= total) return;
    out[i] = agg2[i] + b2[i & 1];
}

// ---------------------------------------------------------------------------
extern "C" void kernel_launch(void* const* d_in, const int* in_sizes, int n_in,
                              void* d_out, int out_size, void* d_ws, size_t ws_size,
                              hipStream_t stream) {
    const float*     x     = (const float*)d_in[0];
    const long long* ei    = (const long long*)d_in[1];   // int64 [2,E]
    const float*     W1    = (const float*)d_in[2];
    const float*     asr1  = (const float*)d_in[3];
    const float*     ads1  = (const float*)d_in[4];
    const float*     b1    = (const float*)d_in[5];
    const float*     W2    = (const float*)d_in[6];
    const float*     asr2  = (const float*)d_in[7];
    const float*     ads2  = (const float*)d_in[8];
    const float*     b2    = (const float*)d_in[9];
    float*           out   = (float*)d_out;

    const int N = in_sizes[0] / 3;
    const int E = in_sizes[1] / 2;
    const long long EP = (long long)E + N;        // edges + self-loops

    // --- workspace carve (256B aligned) ---
    char* p = (char*)d_ws;
    auto carve = [&](size_t bytes) -> char* {
        char* r = p; p += (bytes + 255) & ~(size_t)255; return r;
    };
    int2*     ep   = (int2*)    carve((size_t)EP * 8);
    float*    h1   = (float*)   carve((size_t)N * F1 * 4);
    float*    als1 = (float*)   carve((size_t)N * HEADS * 4);
    float*    ald1 = (float*)   carve((size_t)N * HEADS * 4);
    unsigned* m1   = (unsigned*)carve((size_t)N * HEADS * 4);
    float*    s1   = (float*)   carve((size_t)N * HEADS * 4);
    float*    agg1 = (float*)   carve((size_t)N * F1 * 4);
    float*    t2   = (float*)   carve((size_t)N * 2 * 4);
    float*    als2 = (float*)   carve((size_t)N * 4);
    float*    ald2 = (float*)   carve((size_t)N * 4);
    unsigned* m2   = (unsigned*)carve((size_t)N * 4);
    float*    s2   = (float*)   carve((size_t)N * 4);
    float*    agg2 = (float*)   carve((size_t)N * 2 * 4);

    const int BS = 256;
    const int ntiles  = (N + 15) / 16;
    const int tblocks = (ntiles + 7) / 8;               // 8 waves/block
    const int nblocks = (N + BS - 1) / BS;
    const long long eh1 = EP * HEADS;
    const int pblocks  = (int)((EP + BS - 1) / BS);
    const int e1blocks = (int)((eh1 + BS - 1) / BS);
    const long long no2 = (long long)N * 2;
    const int oblocks  = (int)((no2 + BS - 1) / BS);

    // ---- shared prep ----
    gat_edge_prep  <<<pblocks, BS, 0, stream>>>(ei, E, ep, EP);

    // ---- Layer 1 ----
    gat_l1_xform   <<<tblocks, BS, 0, stream>>>(x, W1, h1, N, ntiles);
    gat_l1_al_init <<<nblocks, BS, 0, stream>>>(h1, asr1, ads1, als1, ald1, m1, s1, agg1, N);
    gat_l1_edge_max<<<pblocks, BS, 0, stream>>>(ep, als1, ald1, m1, EP);
    gat_l1_edge_sum<<<pblocks, BS, 0, stream>>>(ep, als1, ald1, m1, s1, EP);
    gat_l1_edge_agg<<<e1blocks, BS, 0, stream>>>(ep, als1, ald1, m1, s1, h1, agg1, eh1);

    // ---- Layer 2 (bias+ELU fused into xform A-loads) ----
    gat_l2_xform   <<<tblocks, BS, 0, stream>>>(agg1, b1, W2, t2, N, ntiles);
    gat_l2_al_init <<<nblocks, BS, 0, stream>>>(t2, asr2, ads2, als2, ald2, m2, s2, agg2, N);
    gat_l2_edge_max<<<pblocks, BS, 0, stream>>>(ep, als2, ald2, m2, EP);
    gat_l2_edge_sum<<<pblocks, BS, 0, stream>>>(ep, als2, ald2, m2, s2, EP);
    gat_l2_edge_agg<<<pblocks, BS, 0, stream>>>(ep, als2, ald2, m2, s2, t2, agg2, EP);
    gat_l2_fin     <<<oblocks, BS, 0, stream>>>(agg2, b2, out, no2);
}